// GAT_attention_multi_12910671692372
// MI455X (gfx1250) — compile-verified
//
#include <hip/hip_runtime.h>

// ---------------------------------------------------------------------------
// GAT multi-head attention, reformulated to batched GEMMs for CDNA5 WMMA.
//   value[b,i,j,:] = LN(ua_i * ua_j)  is never materialized:
//     mu  = (ua @ ua^T)/64            (WMMA GEMM)
//     m2  = (ua^2 @ (ua^2)^T)/64      (WMMA GEMM, operand squared in regs)
//     out[b,j,d] = ln_w[d]*(ua[j,d]*S1[j,d] - S2[j]) + ln_b[d]*S3[j]
//     S1 = (w*r)^T @ ua               (WMMA GEMM)
// fp32 throughout (reference is fp32) via V_WMMA_F32_16X16X4_F32.
// ---------------------------------------------------------------------------

typedef __attribute__((ext_vector_type(2))) float v2f;
typedef __attribute__((ext_vector_type(8))) float v8f;

#define BB      32      // batch
#define NODES   202
#define NN      200     // attended nodes
#define NP      208     // padded to 13*16 for WMMA tiling
#define DD      64
#define NT      13      // NP/16 tiles
#define LN_EPS  1e-5f
#define SLOPE   0.01f   // jax.nn.leaky_relu default
#define UASTR   65      // LDS row stride (+1 float) to kill bank conflicts

__device__ __forceinline__ float leaky(float x) { return x > 0.f ? x : SLOPE * x; }

__device__ __forceinline__ float wave_sum(float v) {
#pragma unroll
    for (int m = 16; m >= 1; m >>= 1) v += __shfl_xor(v, m, 32);
    return v;
}
__device__ __forceinline__ float wave_max(float v) {
#pragma unroll
    for (int m = 16; m >= 1; m >>= 1) v = fmaxf(v, __shfl_xor(v, m, 32));
    return v;
}

// ---------------- workspace layout (float offsets) ----------------
constexpr size_t OFF_EMBLN  = 0;                                   // B*202*64
constexpr size_t OFF_UA     = OFF_EMBLN  + (size_t)BB * NODES * DD; // B*208*64
constexpr size_t OFF_SQ1    = OFF_UA     + (size_t)BB * NP * DD;
constexpr size_t OFF_SK1    = OFF_SQ1    + (size_t)BB * NP;
constexpr size_t OFF_SQ2    = OFF_SK1    + (size_t)BB * NP;
constexpr size_t OFF_SK2    = OFF_SQ2    + (size_t)BB * NP;
constexpr size_t OFF_SIC    = OFF_SK2    + (size_t)BB * NP;        // 2*B
constexpr size_t OFF_CONSTS = OFF_SIC    + (size_t)2 * BB;         // 6*64 + 8
constexpr size_t OFF_S2     = OFF_CONSTS + (size_t)6 * DD + 8;
constexpr size_t OFF_S3     = OFF_S2     + (size_t)BB * NP;
constexpr size_t OFF_WR     = OFF_S3     + (size_t)BB * NP;        // B*208*208

// ---------------- zero the accumulators / pad rows every launch ----------------
__global__ void k_zero(float* S2, float* S3, float* ua) {
    int t = blockIdx.x * blockDim.x + threadIdx.x;
    if (t < BB * NP) { S2[t] = 0.f; S3[t] = 0.f; }
    if (t < BB * 8 * DD) {               // ua pad rows 200..207 per batch
        int b = t / (8 * DD); int rem = t % (8 * DD);
        int r = NN + rem / DD; int d = rem % DD;
        ua[((size_t)b * NP + r) * DD + d] = 0.f;
    }
}

// ---------------- per-head projection constants ----------------
// vq = aq @ Ww  (so sq[b,i] = ua_i . vq + Wb.aq), etc.
__global__ void k_consts(const float* __restrict__ W1, const float* __restrict__ W1b,
                         const float* __restrict__ W2, const float* __restrict__ W2b,
                         const float* __restrict__ a1, const float* __restrict__ a2,
                         float* __restrict__ consts) {
    int e = threadIdx.x;  // 64 threads
    float vq1 = 0, vk1 = 0, vi1 = 0, vq2 = 0, vk2 = 0, vi2 = 0;
    for (int d = 0; d < DD; ++d) {
        float w1 = W1[d * DD + e], w2 = W2[d * DD + e];
        vq1 += a1[d] * w1;  vk1 += a1[DD + d] * w1;  vi1 += a1[2 * DD + d] * w1;
        vq2 += a2[d] * w2;  vk2 += a2[DD + d] * w2;  vi2 += a2[2 * DD + d] * w2;
    }
    consts[0 * DD + e] = vq1; consts[1 * DD + e] = vk1; consts[2 * DD + e] = vi1;
    consts[3 * DD + e] = vq2; consts[4 * DD + e] = vk2; consts[5 * DD + e] = vi2;
    if (e == 0) {
        float c[6] = {0, 0, 0, 0, 0, 0};
        for (int d = 0; d < DD; ++d) {
            c[0] += W1b[d] * a1[d]; c[1] += W1b[d] * a1[DD + d]; c[2] += W1b[d] * a1[2 * DD + d];
            c[3] += W2b[d] * a2[d]; c[4] += W2b[d] * a2[DD + d]; c[5] += W2b[d] * a2[2 * DD + d];
        }
#pragma unroll
        for (int k = 0; k < 6; ++k) consts[6 * DD + k] = c[k];
    }
}

// ---------------- LayerNorm of all embedding rows (1 wave per row) ----------------
__global__ void k_layernorm(const float* __restrict__ emb, const float* __restrict__ lw,
                            const float* __restrict__ lb, float* __restrict__ embln) {
    int row = blockIdx.x;            // B*NODES
    int lane = threadIdx.x;          // 32 (wave32)
    float2 v = ((const float2*)(emb + (size_t)row * DD))[lane];
    float s = wave_sum(v.x + v.y);
    float ss = wave_sum(v.x * v.x + v.y * v.y);
    float mu = s * (1.f / DD);
    float r = rsqrtf(ss * (1.f / DD) - mu * mu + LN_EPS);
    float2 w = ((const float2*)lw)[lane], b2 = ((const float2*)lb)[lane];
    float2 y; y.x = (v.x - mu) * r * w.x + b2.x; y.y = (v.y - mu) * r * w.y + b2.y;
    ((float2*)(embln + (size_t)row * DD))[lane] = y;
}

// ---------------- ua rows + rank-1 score components (1 wave per (b,i)) ----------------
__global__ void k_prep_rows(const float* __restrict__ embln, const float* __restrict__ consts,
                            float* __restrict__ ua, float* __restrict__ SQ1, float* __restrict__ SK1,
                            float* __restrict__ SQ2, float* __restrict__ SK2) {
    int blk = blockIdx.x;            // B*NN
    int b = blk / NN, i = blk % NN;
    int lane = threadIdx.x;
    float2 u = ((const float2*)(embln + ((size_t)b * NODES + 0) * DD))[lane];
    float2 x = ((const float2*)(embln + ((size_t)b * NODES + 2 + i) * DD))[lane];
    float2 p; p.x = u.x * x.x; p.y = u.y * x.y;
    ((float2*)(ua + ((size_t)b * NP + i) * DD))[lane] = p;
    float2 vq1 = ((const float2*)(consts + 0 * DD))[lane];
    float2 vk1 = ((const float2*)(consts + 1 * DD))[lane];
    float2 vq2 = ((const float2*)(consts + 3 * DD))[lane];
    float2 vk2 = ((const float2*)(consts + 4 * DD))[lane];
    float dq1 = wave_sum(p.x * vq1.x + p.y * vq1.y);
    float dk1 = wave_sum(p.x * vk1.x + p.y * vk1.y);
    float dq2 = wave_sum(p.x * vq2.x + p.y * vq2.y);
    float dk2 = wave_sum(p.x * vk2.x + p.y * vk2.y);
    if (lane == 0) {
        SQ1[b * NP + i] = dq1 + consts[6 * DD + 0];
        SK1[b * NP + i] = dk1 + consts[6 * DD + 1];
        SQ2[b * NP + i] = dq2 + consts[6 * DD + 3];
        SK2[b * NP + i] = dk2 + consts[6 * DD + 4];
    }
}

// ---------------- per-batch: si constants + output row 0 ----------------
__global__ void k_prep_batch(const float* __restrict__ embln, const float* __restrict__ consts,
                             const float* __restrict__ a1b, const float* __restrict__ a2b,
                             float* __restrict__ sic, float* __restrict__ out) {
    int b = blockIdx.x; int lane = threadIdx.x;
    float2 u = ((const float2*)(embln + ((size_t)b * NODES + 0) * DD))[lane];
    float2 ii = ((const float2*)(embln + ((size_t)b * NODES + 1) * DD))[lane];
    float2 o; o.x = leaky(u.x * ii.x); o.y = leaky(u.y * ii.y);
    ((float2*)(out + (size_t)b * (NN + 1) * DD))[lane] = o;     // em_out row 0
    float2 vi1 = ((const float2*)(consts + 2 * DD))[lane];
    float2 vi2 = ((const float2*)(consts + 5 * DD))[lane];
    float d1 = wave_sum(ii.x * vi1.x + ii.y * vi1.y);
    float d2 = wave_sum(ii.x * vi2.x + ii.y * vi2.y);
    if (lane == 0) {
        sic[2 * b + 0] = d1 + consts[6 * DD + 2] + a1b[0];
        sic[2 * b + 1] = d2 + consts[6 * DD + 5] + a2b[0];
    }
}

// ---------------- main: mu/m2 WMMA GEMMs + softmax -> wr, S2, S3 ----------------
// grid (13 i-tiles, B), 256 threads (8 waves). Each block: 16 rows x all 208 cols.
__global__ void __launch_bounds__(256) k_attn(
        const float* __restrict__ ua, const float* __restrict__ SQ1, const float* __restrict__ SK1,
        const float* __restrict__ SQ2, const float* __restrict__ SK2, const float* __restrict__ sic,
        float* __restrict__ wr, float* __restrict__ S2, float* __restrict__ S3) {
    __shared__ float sUA[NP * UASTR];
    __shared__ float sSK1[NN], sSK2[NN];
    __shared__ float sSQ1[16], sSQ2[16];
    __shared__ float sM1[16], sD1[16], sM2[16], sD2[16];

    int b = blockIdx.y, i_base = blockIdx.x * 16;
    int tid = threadIdx.x, lane = tid & 31, wave = tid >> 5;
    const float* uab = ua + (size_t)b * NP * DD;

    for (int idx = tid; idx < NP * DD; idx += 256)
        sUA[(idx / DD) * UASTR + (idx % DD)] = uab[idx];
    for (int j = tid; j < NN; j += 256) { sSK1[j] = SK1[b * NP + j]; sSK2[j] = SK2[b * NP + j]; }
    if (tid < 16) {
        int i = i_base + tid;
        sSQ1[tid] = (i < NN) ? SQ1[b * NP + i] : 0.f;
        sSQ2[tid] = (i < NN) ? SQ2[b * NP + i] : 0.f;
    }
    float si1c = sic[2 * b + 0], si2c = sic[2 * b + 1];
    __syncthreads();

    // softmax stats over j (scores are rank-1; independent of the GEMMs)
#pragma unroll
    for (int t = 0; t < 2; ++t) {
        int m = wave * 2 + t;
        float q1 = sSQ1[m], q2 = sSQ2[m];
        float mx1 = -3.4e38f, mx2 = -3.4e38f;
        for (int j = lane; j < NN; j += 32) {
            mx1 = fmaxf(mx1, leaky(q1 + sSK1[j] + si1c));
            mx2 = fmaxf(mx2, leaky(q2 + sSK2[j] + si2c));
        }
        mx1 = wave_max(mx1); mx2 = wave_max(mx2);
        float d1 = 0.f, d2 = 0.f;
        for (int j = lane; j < NN; j += 32) {
            d1 += __expf(leaky(q1 + sSK1[j] + si1c) - mx1);
            d2 += __expf(leaky(q2 + sSK2[j] + si2c) - mx2);
        }
        d1 = wave_sum(d1); d2 = wave_sum(d2);
        if (lane == 0) { sM1[m] = mx1; sD1[m] = d1; sM2[m] = mx2; sD2[m] = d2; }
    }
    __syncthreads();

    int hi = lane >> 4, ln = lane & 15;
    for (int jt = wave; jt < NT; jt += 8) {
        int j0 = jt * 16;
        v8f accMu = {0, 0, 0, 0, 0, 0, 0, 0};
        v8f accM2 = {0, 0, 0, 0, 0, 0, 0, 0};
#pragma unroll
        for (int k0 = 0; k0 < DD; k0 += 4) {
            int ka = k0 + 2 * hi;
            v2f a, bb;
            a.x = sUA[(i_base + ln) * UASTR + ka];
            a.y = sUA[(i_base + ln) * UASTR + ka + 1];
            bb.x = sUA[(j0 + ln) * UASTR + ka];     // B[k][n] = ua[j0+n][k]
            bb.y = sUA[(j0 + ln) * UASTR + ka + 1];
            accMu = __builtin_amdgcn_wmma_f32_16x16x4_f32(false, a, false, bb,
                                                          (short)0, accMu, false, false);
            v2f a2 = a * a, b2 = bb * bb;           // ua^2 operand built in regs
            accM2 = __builtin_amdgcn_wmma_f32_16x16x4_f32(false, a2, false, b2,
                                                          (short)0, accM2, false, false);
        }
        int j = j0 + ln;
        float s2a = 0.f, s3a = 0.f;
#pragma unroll
        for (int v = 0; v < 8; ++v) {
            int m = v + 8 * hi;
            int ig = i_base + m;
            float mu = accMu[v] * (1.f / DD);
            float m2 = accM2[v] * (1.f / DD);
            float r = rsqrtf(m2 - mu * mu + LN_EPS);
            float wv = 0.f, wrv = 0.f;
            if (ig < NN && j < NN) {
                float a1 = __expf(leaky(sSQ1[m] + sSK1[j] + si1c) - sM1[m]) / sD1[m];
                float a2v = __expf(leaky(sSQ2[m] + sSK2[j] + si2c) - sM2[m]) / sD2[m];
                wv = 0.5f * (a1 + a2v);
                wrv = wv * r;
            }
            wr[((size_t)b * NP + ig) * NP + j] = wrv;  // pads written as exact zeros
            s2a += wrv * mu;
            s3a += wv;
        }
        s2a += __shfl_xor(s2a, 16, 32);   // lanes L and L^16 share column j
        s3a += __shfl_xor(s3a, 16, 32);
        if (hi == 0) {
            atomicAdd(&S2[b * NP + j], s2a);
            atomicAdd(&S3[b * NP + j], s3a);
        }
    }
}

// ---------------- S1 = wr^T @ ua (WMMA) + final combine + leaky ----------------
// grid (13 j-tiles, B), 128 threads (4 waves; wave = d-tile). L2-resident operands.
__global__ void __launch_bounds__(128) k_out(
        const float* __restrict__ ua, const float* __restrict__ wr,
        const float* __restrict__ S2, const float* __restrict__ S3,
        const float* __restrict__ lw, const float* __restrict__ lb, float* __restrict__ out) {
    int b = blockIdx.y, j_base = blockIdx.x * 16;
    int tid = threadIdx.x, lane = tid & 31, wave = tid >> 5;
    int n0 = wave * 16;                 // d-tile
    int hi = lane >> 4, ln = lane & 15;
    const float* uab = ua + (size_t)b * NP * DD;
    const float* wrb = wr + (size_t)b * NP * NP;
    v8f acc = {0, 0, 0, 0, 0, 0, 0, 0};
#pragma unroll 4
    for (int k0 = 0; k0 < NP; k0 += 4) {
        int ka = k0 + 2 * hi;
        v2f a, bb;
        a.x = wrb[(size_t)ka * NP + j_base + ln];        // A[j][i] = wr[i][j]
        a.y = wrb[(size_t)(ka + 1) * NP + j_base + ln];
        bb.x = uab[ka * DD + n0 + ln];                   // B[i][d] = ua[i][d]
        bb.y = uab[(ka + 1) * DD + n0 + ln];
        acc = __builtin_amdgcn_wmma_f32_16x16x4_f32(false, a, false, bb,
                                                    (short)0, acc, false, false);
    }
    int d = n0 + ln;
    float lwv = lw[d], lbv = lb[d];
#pragma unroll
    for (int v = 0; v < 8; ++v) {
        int j = j_base + v + 8 * hi;
        if (j < NN) {
            float val = lwv * (uab[j * DD + d] * acc[v] - S2[b * NP + j]) + lbv * S3[b * NP + j];
            out[((size_t)b * (NN + 1) + 1 + j) * DD + d] = leaky(val);
        }
    }
}

// ---------------------------------------------------------------------------
extern "C" void kernel_launch(void* const* d_in, const int* in_sizes, int n_in,
                              void* d_out, int out_size, void* d_ws, size_t ws_size,
                              hipStream_t stream) {
    (void)in_sizes; (void)n_in; (void)out_size; (void)ws_size;
    const float* emb = (const float*)d_in[0];
    const float* lnw = (const float*)d_in[1];
    const float* lnb = (const float*)d_in[2];
    const float* W1w = (const float*)d_in[3];
    const float* W1b = (const float*)d_in[4];
    const float* W2w = (const float*)d_in[5];
    const float* W2b = (const float*)d_in[6];
    const float* a1w = (const float*)d_in[7];
    const float* a1b = (const float*)d_in[8];
    const float* a2w = (const float*)d_in[9];
    const float* a2b = (const float*)d_in[10];
    float* out = (float*)d_out;
    float* ws  = (float*)d_ws;

    float* embln  = ws + OFF_EMBLN;
    float* ua     = ws + OFF_UA;
    float* SQ1    = ws + OFF_SQ1;
    float* SK1    = ws + OFF_SK1;
    float* SQ2    = ws + OFF_SQ2;
    float* SK2    = ws + OFF_SK2;
    float* sic    = ws + OFF_SIC;
    float* consts = ws + OFF_CONSTS;
    float* S2     = ws + OFF_S2;
    float* S3     = ws + OFF_S3;
    float* wrbuf  = ws + OFF_WR;

    k_zero<<<64, 256, 0, stream>>>(S2, S3, ua);
    k_consts<<<1, 64, 0, stream>>>(W1w, W1b, W2w, W2b, a1w, a2w, consts);
    k_layernorm<<<BB * NODES, 32, 0, stream>>>(emb, lnw, lnb, embln);
    k_prep_rows<<<BB * NN, 32, 0, stream>>>(embln, consts, ua, SQ1, SK1, SQ2, SK2);
    k_prep_batch<<<BB, 32, 0, stream>>>(embln, consts, a1b, a2b, sic, out);
    k_attn<<<dim3(NT, BB), 256, 0, stream>>>(ua, SQ1, SK1, SQ2, SK2, sic, wrbuf, S2, S3);
    k_out<<<dim3(NT, BB), 128, 0, stream>>>(ua, wrbuf, S2, S3, lnw, lnb, out);
}